// BayesianTensorNetwork_34445637714224
// MI455X (gfx1250) — compile-verified
//
#include <hip/hip_runtime.h>

#define D_DIM 784
#define B_DIM 32
#define S_DIM 64
#define O_DIM 10
#define LCH   16                 // chunk length (784 = 49*16)
#define NCH   49                 // chunks
#define CORE_ELEMS (D_DIM * B_DIM * 2 * B_DIM)   // 1,605,632 per-sample eps stride

typedef float v2f __attribute__((ext_vector_type(2)));
typedef float v8f __attribute__((ext_vector_type(8)));

// D = A(16x4 f32) * B(4x16 f32) + C(16x16 f32)
#define WMMA4(A, Bf, C) \
  __builtin_amdgcn_wmma_f32_16x16x4_f32(false, (A), false, (Bf), (short)0, (C), false, false)

// Phase 1: per (sample, chunk) wave computes P = M_{d0} * M_{d0+1} * ... * M_{d0+15}
// (left-to-right, matching carry . M0 . M1 ... order). Running product R lives in
// LDS (pitch 33). M matrices stream into B-fragment REGISTERS via two statically
// named buffers (bA/bB) so the software pipeline never spills to scratch.
__global__ __launch_bounds__(128) void btn_phase1(
    const float* __restrict__ x,
    const float* __restrict__ cm,    // core_means   (D,B,2,B)
    const float* __restrict__ clv,   // core_logvars (D,B,2,B)
    const float* __restrict__ eps,   // eps_cores  (S,D,B,2,B)
    float* __restrict__ P)           // (S,NCH,32,32) chunk products
{
  __shared__ float Rb[4][B_DIM * 33];
  const int wave  = threadIdx.x >> 5;
  const int lane  = threadIdx.x & 31;
  const int lid   = lane & 15;
  const int lhalf = lane >> 4;
  const int gw = blockIdx.x * 4 + wave;      // 0..3135
  const int s  = gw / NCH;
  const int c  = gw % NCH;
  const int d0 = c * LCH;
  float* R = Rb[wave];
  const size_t soff = (size_t)s * CORE_ELEMS;

  // Seed R with the selected matrix at d0 (row-major, pitch 33).
  {
    const int bit = (x[d0] > 0.5f) ? 1 : 0;
    const size_t base = (size_t)d0 * 2048 + (size_t)bit * 32;  // idx(d,j,b,k)=d*2048+j*64+b*32+k
    #pragma unroll 4
    for (int r = 0; r < 32; ++r) {
      const size_t idx = base + (size_t)r * 64 + lane;
      R[r * 33 + lane] = cm[idx] + eps[soff + idx] * __expf(0.5f * clv[idx]);
    }
  }

  // Two statically-named fragment buffers (all indices compile-time constant
  // after unrolling -> SROA keeps them in VGPRs, no scratch).
  v2f bA[8][2], bB[8][2];

  auto loadB = [&](int d, v2f (&bf)[8][2]) {
    const int bit = (x[d] > 0.5f) ? 1 : 0;
    const size_t base = (size_t)d * 2048 + (size_t)bit * 32;
    #pragma unroll
    for (int jt = 0; jt < 8; ++jt) {
      #pragma unroll
      for (int ct = 0; ct < 2; ++ct) {
        #pragma unroll
        for (int v = 0; v < 2; ++v) {
          // B tile layout: N = lane%16, K = 2*(lane/16) + v
          const int r  = 4 * jt + v + 2 * lhalf;   // row of M (contracted dim j)
          const int cc = 16 * ct + lid;            // col of M (output dim k)
          const size_t idx = base + (size_t)r * 64 + cc;
          bf[jt][ct][v] = cm[idx] + eps[soff + idx] * __expf(0.5f * clv[idx]);
        }
      }
    }
  };

  // One product step: R <- R * M, with M's B-fragments in bf.
  auto step = [&](v2f (&bf)[8][2]) {
    v8f acc[2][2];
    #pragma unroll
    for (int at = 0; at < 2; ++at)
      #pragma unroll
      for (int ct = 0; ct < 2; ++ct)
        acc[at][ct] = (v8f){0.f, 0.f, 0.f, 0.f, 0.f, 0.f, 0.f, 0.f};

    // C[at][ct] += sum_jt A(R)[at][jt] * B(M)[jt][ct] ; 32 WMMAs per step
    #pragma unroll
    for (int jt = 0; jt < 8; ++jt) {
      const int colb = 4 * jt + 2 * lhalf;     // A layout: M=lane%16, K=2*(lane/16)+v
      v2f a0, a1;
      a0.x = R[lid * 33 + colb];
      a0.y = R[lid * 33 + colb + 1];
      a1.x = R[(16 + lid) * 33 + colb];
      a1.y = R[(16 + lid) * 33 + colb + 1];
      acc[0][0] = WMMA4(a0, bf[jt][0], acc[0][0]);
      acc[0][1] = WMMA4(a0, bf[jt][1], acc[0][1]);
      acc[1][0] = WMMA4(a1, bf[jt][0], acc[1][0]);
      acc[1][1] = WMMA4(a1, bf[jt][1], acc[1][1]);
    }

    // Write D tiles back to R. C/D layout: M = v + 8*(lane/16), N = lane%16.
    #pragma unroll
    for (int at = 0; at < 2; ++at)
      #pragma unroll
      for (int ct = 0; ct < 2; ++ct)
        #pragma unroll
        for (int v = 0; v < 8; ++v)
          R[(16 * at + v + 8 * lhalf) * 33 + 16 * ct + lid] = acc[at][ct][v];
  };

  // Software pipeline over 15 steps (matrices d0+1 .. d0+15), ping-pong bA/bB:
  loadB(d0 + 1, bA);
  for (int ii = 0; ii < 7; ++ii) {
    const int i0 = 1 + 2 * ii;
    loadB(d0 + i0 + 1, bB);   // prefetch next matrix while...
    step(bA);                 // ...multiplying with the current one
    loadB(d0 + i0 + 2, bA);
    step(bB);
  }
  step(bA);                   // step 15 (matrix d0+15)

  float* outp = P + ((size_t)(s * NCH + c) << 10);
  #pragma unroll 4
  for (int r = 0; r < 32; ++r)
    outp[r * 32 + lane] = R[r * 33 + lane];
}

// Phase 2: per-sample, chain carry through the 49 chunk products, then project
// through right = right_mean + eps_right * exp(0.5*right_logvar).
__global__ __launch_bounds__(32) void btn_phase2(
    const float* __restrict__ P,
    const float* __restrict__ lm,  const float* __restrict__ llv,
    const float* __restrict__ rm,  const float* __restrict__ rlv,
    const float* __restrict__ el,  const float* __restrict__ er,
    float* __restrict__ out)
{
  __shared__ float sc[B_DIM];
  const int s = blockIdx.x;
  const int lane = threadIdx.x;
  float carry = lm[lane] + el[s * B_DIM + lane] * __expf(0.5f * llv[lane]);
  for (int c = 0; c < NCH; ++c) {
    sc[lane] = carry;
    __syncthreads();
    const float* Pc = P + ((size_t)(s * NCH + c) << 10);
    float a = 0.f;
    #pragma unroll 8
    for (int j = 0; j < B_DIM; ++j)
      a = __builtin_fmaf(sc[j], Pc[j * B_DIM + lane], a);
    __syncthreads();
    carry = a;
  }
  sc[lane] = carry;
  __syncthreads();
  if (lane < O_DIM) {
    float a = 0.f;
    #pragma unroll
    for (int j = 0; j < B_DIM; ++j) {
      const int ro = j * O_DIM + lane;
      const float rv = rm[ro] + er[(size_t)s * (B_DIM * O_DIM) + ro] * __expf(0.5f * rlv[ro]);
      a = __builtin_fmaf(sc[j], rv, a);
    }
    out[s * O_DIM + lane] = a;
  }
}

extern "C" void kernel_launch(void* const* d_in, const int* in_sizes, int n_in,
                              void* d_out, int out_size, void* d_ws, size_t ws_size,
                              hipStream_t stream) {
  const float* x   = (const float*)d_in[0];
  const float* cm  = (const float*)d_in[1];
  const float* clv = (const float*)d_in[2];
  const float* lm  = (const float*)d_in[3];
  const float* llv = (const float*)d_in[4];
  const float* rm  = (const float*)d_in[5];
  const float* rlv = (const float*)d_in[6];
  const float* ec  = (const float*)d_in[7];
  const float* el  = (const float*)d_in[8];
  const float* er  = (const float*)d_in[9];
  float* P = (float*)d_ws;   // needs S*NCH*1024*4 = 12,845,056 bytes of scratch

  btn_phase1<<<(S_DIM * NCH) / 4, 128, 0, stream>>>(x, cm, clv, ec, P);
  btn_phase2<<<S_DIM, 32, 0, stream>>>(P, lm, llv, rm, rlv, el, er, (float*)d_out);
}